// OccupancyCriterion_32650341384710
// MI455X (gfx1250) — compile-verified
//
#include <hip/hip_runtime.h>
#include <cstdint>

// Problem geometry (fixed by the reference: B=32, C=1, H=W=512)
#define HW    512
#define NPI   (HW * HW)          // 262144 elements per image (2^18)
#define NIMG  32
#define MTOT  (NIMG * NPI)       // 8388608 total elements (2^23)

// Bitonic LDS tile: 32768 u32 = 128 KB dynamic LDS (CDNA5 WGP has 320 KB)
#define TILE  32768
#define TPB   1024
#define PAIRS (TILE / 2 / TPB)   // 16 compare-exchange pairs per thread/substage
#define VECS  (TILE / 4 / TPB)   // 8 x 16-byte chunks per thread for fills

// Loss weights (OccupancyCriterion defaults)
#define W_ALPHA 0.25f
#define W_POSW  20.0f
#define W_LL 1.0f
#define W_LF 0.5f
#define W_LE 0.3f
#define W_LB 0.2f

// ---------------------------------------------------------------------------
// helpers
// ---------------------------------------------------------------------------

__device__ __forceinline__ float softplusf(float x) {
  // stable log(1+e^x)
  return fmaxf(x, 0.0f) + log1pf(expf(-fabsf(x)));
}

// Encode error e and label gt into one u32 such that ascending u32 order ==
// descending error order.  Bit 0 carries the label (costs <=1 ulp of e).
__device__ __forceinline__ unsigned encode_item(float e, unsigned gt) {
  unsigned b = __float_as_uint(e);
  unsigned m = (b & 0x80000000u) ? ~b : (b | 0x80000000u); // monotonic asc map
  unsigned key = ~m;                                       // asc key = desc e
  return (key & ~1u) | gt;
}
__device__ __forceinline__ float decode_err(unsigned item) {
  unsigned m = ~(item & ~1u);
  unsigned fb = (m & 0x80000000u) ? (m & 0x7FFFFFFFu) : ~m;
  return __uint_as_float(fb);
}

// Copy TILE u32 (128KB) global tile -> LDS via the gfx1250 async global->LDS
// engine (ASYNCcnt path); falls back to plain vector copies if the builtin is
// unavailable.  Probe-derived prototype: (int4 AS1*, int4 AS3*, imm, imm).
typedef int v4i_async __attribute__((vector_size(4 * sizeof(int))));

__device__ __forceinline__ void fill_lds_tile(unsigned* __restrict__ s,
                                              const unsigned* __restrict__ g,
                                              int tid) {
#if defined(__gfx1250__) && __has_builtin(__builtin_amdgcn_global_load_async_to_lds_b128)
#pragma unroll
  for (int k = 0; k < VECS; ++k) {
    __builtin_amdgcn_global_load_async_to_lds_b128(
        (__attribute__((address_space(1))) v4i_async*)(g + k * (TPB * 4) + tid * 4),
        (__attribute__((address_space(3))) v4i_async*)(s + k * (TPB * 4) + tid * 4),
        0, 0);
  }
#if __has_builtin(__builtin_amdgcn_s_wait_asynccnt)
  __builtin_amdgcn_s_wait_asynccnt(0);
#else
  asm volatile("s_wait_asynccnt 0" ::: "memory");
#endif
#else
#pragma unroll
  for (int k = 0; k < VECS; ++k) {
    uint4 v = *(const uint4*)(g + k * (TPB * 4) + tid * 4);
    *(uint4*)(s + k * (TPB * 4) + tid * 4) = v;
  }
#endif
}

__device__ __forceinline__ void store_tile(const unsigned* __restrict__ s,
                                           unsigned* __restrict__ g, int tid) {
#pragma unroll
  for (int k = 0; k < VECS; ++k) {
    uint4 v = *(const uint4*)(s + k * (TPB * 4) + tid * 4);
    *(uint4*)(g + k * (TPB * 4) + tid * 4) = v;
  }
}

// ---------------------------------------------------------------------------
// kernels (bitonic first so the disasm snippet shows the async-LDS hot loop)
// ---------------------------------------------------------------------------

// Fused local sort of a 32768-item tile: the entire bitonic network for
// sizes 2..32768 runs out of LDS with one global read + one global write.
__global__ __launch_bounds__(TPB) void bitonic_local_sort(unsigned* items) {
  extern __shared__ unsigned s[];
  const int tid = threadIdx.x;
  const long long base = (long long)blockIdx.x * TILE;
  fill_lds_tile(s, items + base, tid);
  __syncthreads();
  for (unsigned size = 2; size <= TILE; size <<= 1) {
    for (unsigned stride = size >> 1; stride > 0; stride >>= 1) {
#pragma unroll
      for (int k = 0; k < PAIRS; ++k) {
        unsigned p = (unsigned)tid + (unsigned)k * TPB;
        unsigned i = ((p & ~(stride - 1u)) << 1) | (p & (stride - 1u));
        unsigned j = i + stride;
        unsigned si = ((unsigned)(base + i)) & (NPI - 1u); // segment-local
        bool asc = ((si & size) == 0u);
        unsigned a = s[i], b = s[j];
        if ((a > b) == asc) { s[i] = b; s[j] = a; }
      }
      __syncthreads();
    }
  }
  store_tile(s, items + base, tid);
}

// Fused local merge: strides TILE/2..1 for a given size, in LDS.
__global__ __launch_bounds__(TPB) void bitonic_local_merge(unsigned* items,
                                                           unsigned size) {
  extern __shared__ unsigned s[];
  const int tid = threadIdx.x;
  const long long base = (long long)blockIdx.x * TILE;
  fill_lds_tile(s, items + base, tid);
  __syncthreads();
  for (unsigned stride = TILE / 2; stride > 0; stride >>= 1) {
#pragma unroll
    for (int k = 0; k < PAIRS; ++k) {
      unsigned p = (unsigned)tid + (unsigned)k * TPB;
      unsigned i = ((p & ~(stride - 1u)) << 1) | (p & (stride - 1u));
      unsigned j = i + stride;
      unsigned si = ((unsigned)(base + i)) & (NPI - 1u);
      bool asc = ((si & size) == 0u);
      unsigned a = s[i], b = s[j];
      if ((a > b) == asc) { s[i] = b; s[j] = a; }
    }
    __syncthreads();
  }
  store_tile(s, items + base, tid);
}

// One global compare-exchange substage (stride >= TILE).
__global__ __launch_bounds__(256) void bitonic_global_pass(unsigned* items,
                                                           unsigned size,
                                                           unsigned stride) {
  unsigned p = blockIdx.x * 256u + threadIdx.x; // MTOT/2 pairs
  unsigned i = ((p & ~(stride - 1u)) << 1) | (p & (stride - 1u));
  unsigned j = i + stride;
  unsigned si = i & (NPI - 1u);
  bool asc = ((si & size) == 0u);
  unsigned a = items[i], b = items[j];
  if ((a > b) == asc) { items[i] = b; items[j] = a; }
}

__global__ void init_pass(unsigned* posCnt) {
  if (threadIdx.x < NIMG) posCnt[threadIdx.x] = 0u;
}

// Elementwise pass: focal + weighted-BCE partial sums, per-image positive
// counts, and the packed lovasz sort items.  2048 blocks x 256 thr x 16 items.
__global__ __launch_bounds__(256) void elementwise_pass(
    const float* __restrict__ logits, const float* __restrict__ targets,
    unsigned* __restrict__ items, unsigned* __restrict__ posCnt,
    float* __restrict__ pFocal, float* __restrict__ pBce) {
  __shared__ float sf[256];
  __shared__ unsigned su[256];
  const int tid = threadIdx.x;
  const long long base = (long long)blockIdx.x * 4096;
  float accF = 0.f, accB = 0.f;
  unsigned pos = 0;
#pragma unroll 4
  for (int it = 0; it < 16; ++it) {
    long long idx = base + it * 256 + tid;
    float x = logits[idx];
    float t = targets[idx];
    float spn = softplusf(-x), spp = softplusf(x);
    float bce = t * spn + (1.f - t) * spp;
    float p = 1.f / (1.f + expf(-x));
    float pt = t * p + (1.f - t) * (1.f - p);
    float at = t * W_ALPHA + (1.f - t) * (1.f - W_ALPHA);
    float q = 1.f - pt;
    accF += at * q * q * bce;                   // GAMMA == 2
    accB += W_POSW * t * spn + (1.f - t) * spp; // pos-weighted BCE
    unsigned gt = (t > 0.5f) ? 1u : 0u;
    float e = 1.f - x * (2.f * t - 1.f);        // hinge error
    items[idx] = encode_item(e, gt);
    pos += gt;
  }
  sf[tid] = accF; __syncthreads();
  for (int s = 128; s > 0; s >>= 1) { if (tid < s) sf[tid] += sf[tid + s]; __syncthreads(); }
  if (tid == 0) pFocal[blockIdx.x] = sf[0];
  __syncthreads();
  sf[tid] = accB; __syncthreads();
  for (int s = 128; s > 0; s >>= 1) { if (tid < s) sf[tid] += sf[tid + s]; __syncthreads(); }
  if (tid == 0) pBce[blockIdx.x] = sf[0];
  su[tid] = pos; __syncthreads();
  for (int s = 128; s > 0; s >>= 1) { if (tid < s) su[tid] += su[tid + s]; __syncthreads(); }
  if (tid == 0) atomicAdd(&posCnt[(unsigned)(base / NPI)], su[0]); // int atomic: deterministic
}

// Sobel boundary L1: 16x16 output tile + halo in LDS, zero SAME padding.
__global__ __launch_bounds__(256) void sobel_pass(const float* __restrict__ logits,
                                                  const float* __restrict__ targets,
                                                  float* __restrict__ pBnd) {
  __shared__ float sp[18][20];
  __shared__ float st[18][20];
  __shared__ float sr[256];
  const int tid = threadIdx.x;
  const int ox = blockIdx.x * 16, oy = blockIdx.y * 16, img = blockIdx.z;
  const long long ibase = (long long)img * NPI;
  for (int idx = tid; idx < 18 * 18; idx += 256) {
    int ly = idx / 18, lx = idx % 18;
    int gy = oy + ly - 1, gx = ox + lx - 1;
    float pv = 0.f, tv = 0.f;
    if (gx >= 0 && gx < HW && gy >= 0 && gy < HW) {
      long long o = ibase + (long long)gy * HW + gx;
      float x = logits[o];
      pv = 1.f / (1.f + expf(-x));
      tv = targets[o];
    }
    sp[ly][lx] = pv;
    st[ly][lx] = tv;
  }
  __syncthreads();
  const int tx = tid & 15, ty = tid >> 4;
  const int y = ty + 1, x = tx + 1;
  float a00 = sp[y-1][x-1], a01 = sp[y-1][x], a02 = sp[y-1][x+1];
  float a10 = sp[y  ][x-1],                   a12 = sp[y  ][x+1];
  float a20 = sp[y+1][x-1], a21 = sp[y+1][x], a22 = sp[y+1][x+1];
  float ex = (a02 - a00) + 2.f * (a12 - a10) + (a22 - a20);
  float ey = (a20 - a00) + 2.f * (a21 - a01) + (a22 - a02);
  float magp = sqrtf(ex * ex + ey * ey + 1e-8f);
  float b00 = st[y-1][x-1], b01 = st[y-1][x], b02 = st[y-1][x+1];
  float b10 = st[y  ][x-1],                   b12 = st[y  ][x+1];
  float b20 = st[y+1][x-1], b21 = st[y+1][x], b22 = st[y+1][x+1];
  float fx = (b02 - b00) + 2.f * (b12 - b10) + (b22 - b20);
  float fy = (b20 - b00) + 2.f * (b21 - b01) + (b22 - b02);
  float magt = sqrtf(fx * fx + fy * fy + 1e-8f);
  sr[tid] = fabsf(magp - magt);
  __syncthreads();
  for (int s = 128; s > 0; s >>= 1) { if (tid < s) sr[tid] += sr[tid + s]; __syncthreads(); }
  if (tid == 0) pBnd[(blockIdx.z * 32 + blockIdx.y) * 32 + blockIdx.x] = sr[0];
}

// Lovasz: per-image streaming scan over the sorted items (one WG per image).
// Rank i + inclusive positive prefix c_i give grad directly; no gather needed.
__global__ __launch_bounds__(1024) void lovasz_pass(const unsigned* __restrict__ items,
                                                    const unsigned* __restrict__ posCnt,
                                                    float* __restrict__ lovImg) {
  __shared__ unsigned swsum[32];
  __shared__ unsigned sCarry;
  __shared__ float sred[1024];
  const int tid = threadIdx.x;
  const int img = blockIdx.x;
  const long long base = (long long)img * NPI;
  const float G = (float)posCnt[img];
  const int lane = tid & 31, wv = tid >> 5;
  if (tid == 0) sCarry = 0u;
  __syncthreads();
  float acc = 0.f;
  for (int c = 0; c < NPI / 1024; ++c) {
    unsigned item = items[base + (long long)c * 1024 + tid];
    unsigned gt = item & 1u;
    unsigned mask = (unsigned)__ballot(gt != 0u);            // wave32
    unsigned incl = __popc(mask & (0xFFFFFFFFu >> (31 - lane)));
    if (lane == 31) swsum[wv] = incl;                        // wave total
    __syncthreads();
    if (tid < 32) {                                          // scan 32 wave sums
      unsigned v = swsum[tid];
      for (int off = 1; off < 32; off <<= 1) {
        unsigned n = __shfl_up(v, off, 32);
        if (tid >= off) v += n;
      }
      swsum[tid] = v;
    }
    __syncthreads();
    unsigned carry = sCarry;
    unsigned ci = carry + (wv ? swsum[wv - 1] : 0u) + incl;  // inclusive prefix
    int i = c * 1024 + tid;                                  // global rank
    float e = decode_err(item);
    if (e > 0.f) {                                           // relu gate
      float fi = (float)i, fc = (float)ci;
      float j1 = 1.f - (G - fc) / (G + (fi + 1.f) - fc);
      float contrib;
      if (i == 0) {
        contrib = e * j1;
      } else {
        float fcp = fc - (float)gt;
        float j0 = 1.f - (G - fcp) / (G + fi - fcp);
        contrib = e * (j1 - j0);
      }
      acc += contrib;
    }
    __syncthreads();
    if (tid == 0) sCarry = carry + swsum[31];
    __syncthreads();
  }
  sred[tid] = acc; __syncthreads();
  for (int s = 512; s > 0; s >>= 1) { if (tid < s) sred[tid] += sred[tid + s]; __syncthreads(); }
  if (tid == 0) lovImg[img] = sred[0];
}

// Deterministic final reduction of all partials -> 5 output scalars.
__global__ __launch_bounds__(256) void finalize_pass(const float* __restrict__ pF,
                                                     const float* __restrict__ pB,
                                                     const float* __restrict__ pE,
                                                     const float* __restrict__ pL,
                                                     float* __restrict__ out) {
  __shared__ double sd[256];
  const int tid = threadIdx.x;
  double a;
  a = 0; for (int i = tid; i < 2048; i += 256) a += (double)pF[i];
  sd[tid] = a; __syncthreads();
  for (int s = 128; s > 0; s >>= 1) { if (tid < s) sd[tid] += sd[tid + s]; __syncthreads(); }
  double focal = sd[0] / (double)MTOT; __syncthreads();
  a = 0; for (int i = tid; i < 2048; i += 256) a += (double)pB[i];
  sd[tid] = a; __syncthreads();
  for (int s = 128; s > 0; s >>= 1) { if (tid < s) sd[tid] += sd[tid + s]; __syncthreads(); }
  double bce = sd[0] / (double)MTOT; __syncthreads();
  a = 0; for (int i = tid; i < 32768; i += 256) a += (double)pE[i];
  sd[tid] = a; __syncthreads();
  for (int s = 128; s > 0; s >>= 1) { if (tid < s) sd[tid] += sd[tid + s]; __syncthreads(); }
  double bnd = sd[0] / (double)MTOT; __syncthreads();
  a = 0; for (int i = tid; i < 32; i += 256) a += (double)pL[i];
  sd[tid] = a; __syncthreads();
  for (int s = 128; s > 0; s >>= 1) { if (tid < s) sd[tid] += sd[tid + s]; __syncthreads(); }
  double lov = sd[0] / (double)NIMG;
  if (tid == 0) {
    out[0] = (float)((double)W_LL * lov + (double)W_LF * focal +
                     (double)W_LE * bnd + (double)W_LB * bce);
    out[1] = (float)lov;
    out[2] = (float)focal;
    out[3] = (float)bnd;
    out[4] = (float)bce;
  }
}

// ---------------------------------------------------------------------------
// launcher  (needs ~34 MB of d_ws: 32 MB sort items + partial-sum arrays)
// ---------------------------------------------------------------------------
extern "C" void kernel_launch(void* const* d_in, const int* in_sizes, int n_in,
                              void* d_out, int out_size, void* d_ws, size_t ws_size,
                              hipStream_t stream) {
  (void)in_sizes; (void)n_in; (void)out_size; (void)ws_size;
  const float* logits  = (const float*)d_in[0];
  const float* targets = (const float*)d_in[1];
  float* out = (float*)d_out;

  unsigned char* wsb = (unsigned char*)d_ws;
  unsigned* items  = (unsigned*)wsb;                                   // 32 MB
  unsigned* posCnt = (unsigned*)(wsb + (size_t)MTOT * 4);              // 128 B
  float* pF = (float*)(wsb + (size_t)MTOT * 4 + 128);                  // 2048
  float* pB = pF + 2048;                                               // 2048
  float* pE = pB + 2048;                                               // 32768
  float* pL = pE + 32768;                                              // 32

  const size_t ldsBytes = (size_t)TILE * 4;  // 128 KB dynamic LDS per WG

  init_pass<<<1, 32, 0, stream>>>(posCnt);
  elementwise_pass<<<MTOT / 4096, 256, 0, stream>>>(logits, targets, items,
                                                    posCnt, pF, pB);
  // segmented bitonic sort (per image, 2^18 each), in place:
  // 1 fused local sort + 6 global passes + 3 fused local merges
  bitonic_local_sort<<<MTOT / TILE, TPB, ldsBytes, stream>>>(items);
  for (unsigned size = TILE * 2; size <= (unsigned)NPI; size <<= 1) {
    for (unsigned stride = size >> 1; stride >= (unsigned)TILE; stride >>= 1) {
      bitonic_global_pass<<<MTOT / 2 / 256, 256, 0, stream>>>(items, size, stride);
    }
    bitonic_local_merge<<<MTOT / TILE, TPB, ldsBytes, stream>>>(items, size);
  }
  sobel_pass<<<dim3(HW / 16, HW / 16, NIMG), 256, 0, stream>>>(logits, targets, pE);
  lovasz_pass<<<NIMG, 1024, 0, stream>>>(items, posCnt, pL);
  finalize_pass<<<1, 256, 0, stream>>>(pF, pB, pE, pL, out);
}